// TranslatorEncoderRNN_34368328303083
// MI455X (gfx1250) — compile-verified
//
#include <hip/hip_runtime.h>
#include <math.h>

#define B_  64
#define T_  512
#define E_  256
#define H_  512
#define NWG 128
#define NTHREADS 256

typedef float v2f __attribute__((ext_vector_type(2)));
typedef float v8f __attribute__((ext_vector_type(8)));

__device__ __forceinline__ float sigmoidf_(float x) {
    return 1.0f / (1.0f + __expf(-x));
}

// ---------------------------------------------------------------------------
// Init: zero h/c state + barrier counter, emit mask output.
// ---------------------------------------------------------------------------
__global__ void lstm_init(const int* __restrict__ lengths,
                          float* __restrict__ mask_out,
                          float* __restrict__ h0, float* __restrict__ h1,
                          float* __restrict__ c, unsigned* __restrict__ bar) {
    int idx = blockIdx.x * blockDim.x + threadIdx.x;
    if (idx == 0) *bar = 0u;
    if (idx < B_ * H_) { h0[idx] = 0.0f; h1[idx] = 0.0f; c[idx] = 0.0f; }
    if (idx < B_ * T_) {
        int b = idx / T_;
        int t = idx - b * T_;
        mask_out[idx] = (t < lengths[b]) ? 1.0f : 0.0f;
    }
}

// ---------------------------------------------------------------------------
// Persistent LSTM scan.
// Grid: 128 WGs = 4 M-tiles (batch rows) x 32 col-blocks (h columns).
// Block: 256 threads = 8 wave32 = 4 gate-quarters x 2 K-halves.
// Per step: h-tile (32KB) + gathered x-tile (16KB) staged to LDS via
// GLOBAL_LOAD_ASYNC_TO_LDS_B128 (ASYNCcnt); each wave then runs two
// interleaved v_wmma_f32_16x16x4_f32 accumulate chains (A from LDS,
// B = weights from L2), halves reduced via LDS; elementwise LSTM cell;
// monotonic atomic-counter grid barrier between steps.
// ---------------------------------------------------------------------------
__global__ __launch_bounds__(NTHREADS, 1)
void lstm_scan(const int*   __restrict__ inputs,
               const int*   __restrict__ lengths,
               const int*   __restrict__ resets,
               const float* __restrict__ emb,
               const float* __restrict__ W_ih,
               const float* __restrict__ W_hh,
               const float* __restrict__ b_ih,
               const float* __restrict__ b_hh,
               float* __restrict__ out,
               float* __restrict__ out_hT,
               float* __restrict__ out_cT,
               float* __restrict__ hbuf0,
               float* __restrict__ hbuf1,
               float* __restrict__ cbuf,
               unsigned* __restrict__ bar) {
    __shared__ float hs[16][H_];        // this WG's 16 h rows (all 512 cols)
    __shared__ float xs[16][E_];        // gathered embedding rows
    __shared__ float gtile[8][16][16];  // per-wave partial gate tiles
    __shared__ int   tok[16];

    const int tid  = threadIdx.x;
    const int lane = tid & 31;
    const int wave = tid >> 5;
    const int q    = wave >> 1;               // gate quarter: 0=i 1=f 2=g 3=o
    const int half = wave & 1;                // K-split half
    const int mt   = blockIdx.x & 3;          // batch-row tile (M)
    const int jb   = (blockIdx.x >> 2) * 16;  // h-column block
    const int mbase = mt * 16;
    const int lrow = lane & 15;               // M row (A) / N col (B,C,D)
    const int khi  = lane >> 4;
    const int koff = khi * 2;

    const int gcol = q * H_ + jb + lrow;      // gate column this lane feeds
    const float* Wih_row = W_ih + (size_t)gcol * E_;
    const float* Whh_row = W_hh + (size_t)gcol * H_;
    const float lbias = (half == 0) ? (b_ih[gcol] + b_hh[gcol]) : 0.0f;

    const unsigned hs_base = (unsigned)(uintptr_t)&hs[0][0];
    const unsigned xs_base = (unsigned)(uintptr_t)&xs[0][0];

    for (int t = 0; t < T_; ++t) {
        const float* hcur = (t & 1) ? hbuf1 : hbuf0;
        float*       hnxt = (t & 1) ? hbuf0 : hbuf1;

        __syncthreads();   // previous step done reading hs/xs/tok
        if (tid < 16) tok[tid] = inputs[(mbase + tid) * T_ + t];
        __syncthreads();

        // ---- async stage h tile: rows mbase..mbase+15 are contiguous 32KB
        {
            const float* hsrc = hcur + (size_t)mbase * H_;
#pragma unroll
            for (int i = 0; i < 8; ++i) {
                const int idx = tid + i * NTHREADS;          // dword4 index
                const float* g = hsrc + (size_t)idx * 4;
                const unsigned l = hs_base + (unsigned)idx * 16u;
                asm volatile("global_load_async_to_lds_b128 %0, %1, off"
                             :: "v"(l), "v"(g) : "memory");
            }
        }
        // ---- async stage x tile: 16 gathered embedding rows (1KB each)
        {
#pragma unroll
            for (int i = 0; i < 4; ++i) {
                const int idx = tid + i * NTHREADS;          // dword4 index
                const int row = idx >> 6;
                const int w   = idx & 63;
                const float* g = emb + (size_t)tok[row] * E_ + (size_t)w * 4;
                const unsigned l = xs_base + (unsigned)idx * 16u;
                asm volatile("global_load_async_to_lds_b128 %0, %1, off"
                             :: "v"(l), "v"(g) : "memory");
            }
        }
        if (t + 1 < T_)
            __builtin_prefetch(&inputs[(mbase + lrow) * T_ + t + 1], 0, 1);
        asm volatile("s_wait_asynccnt 0x0" ::: "memory");
        __syncthreads();   // LDS tiles visible to all waves

        // ---- GEMM: two independent accumulate chains per wave
        v8f acc0, acc1;
#pragma unroll
        for (int r = 0; r < 8; ++r) { acc0[r] = lbias; acc1[r] = 0.0f; }

        // x-part: K in [half*128, half*128+128)
        {
            const int k0 = half * (E_ / 2);
#pragma unroll 2
            for (int kk = 0; kk < E_ / 2; kk += 8) {
                const int k = k0 + kk + koff;
                v2f a0 = *(const v2f*)&xs[lrow][k];
                v2f b0 = *(const v2f*)(Wih_row + k);
                acc0 = __builtin_amdgcn_wmma_f32_16x16x4_f32(
                    false, a0, false, b0, (short)0, acc0, false, false);
                v2f a1 = *(const v2f*)&xs[lrow][k + 4];
                v2f b1 = *(const v2f*)(Wih_row + k + 4);
                acc1 = __builtin_amdgcn_wmma_f32_16x16x4_f32(
                    false, a1, false, b1, (short)0, acc1, false, false);
            }
        }
        // h-part: K in [half*256, half*256+256)
        {
            const int k0 = half * (H_ / 2);
#pragma unroll 2
            for (int kk = 0; kk < H_ / 2; kk += 8) {
                const int k = k0 + kk + koff;
                v2f a0 = *(const v2f*)&hs[lrow][k];
                v2f b0 = *(const v2f*)(Whh_row + k);
                acc0 = __builtin_amdgcn_wmma_f32_16x16x4_f32(
                    false, a0, false, b0, (short)0, acc0, false, false);
                v2f a1 = *(const v2f*)&hs[lrow][k + 4];
                v2f b1 = *(const v2f*)(Whh_row + k + 4);
                acc1 = __builtin_amdgcn_wmma_f32_16x16x4_f32(
                    false, a1, false, b1, (short)0, acc1, false, false);
            }
        }

        // ---- spill combined tile: C/D layout -> M = r + 8*khi, N = lane&15
#pragma unroll
        for (int r = 0; r < 8; ++r)
            gtile[wave][r + khi * 8][lrow] = acc0[r] + acc1[r];
        __syncthreads();

        // ---- elementwise LSTM cell on the 16x16 tile (1 elem/thread)
        {
            const int rl  = tid >> 4;
            const int col = tid & 15;
            const int b   = mbase + rl;
            const int j   = jb + col;
            const float gi = gtile[0][rl][col] + gtile[1][rl][col];
            const float gf = gtile[2][rl][col] + gtile[3][rl][col];
            const float gg = gtile[4][rl][col] + gtile[5][rl][col];
            const float go = gtile[6][rl][col] + gtile[7][rl][col];
            const float i_ = sigmoidf_(gi);
            const float f_ = sigmoidf_(gf);
            const float g_ = tanhf(gg);
            const float o_ = sigmoidf_(go);
            const float c_old = cbuf[b * H_ + j];
            const float h_old = hs[rl][j];            // staged copy of hcur
            const float c_new = f_ * c_old + i_ * g_;
            const float h_new = o_ * tanhf(c_new);
            const float m = (t < lengths[b]) ? 1.0f : 0.0f;
            const float r = (float)resets[b * T_ + t];
            const float h_nx = h_new * m + h_old * (1.0f - m);
            const float c_nx = c_new * m + c_old * (1.0f - m);
            out[((size_t)b * T_ + t) * H_ + j] = h_nx;
            const float hc = h_nx * (1.0f - r);
            const float cc = c_nx * (1.0f - r);
            hnxt[b * H_ + j] = hc;
            cbuf[b * H_ + j] = cc;
            if (t == T_ - 1) {
                out_hT[b * H_ + j] = hc;
                out_cT[b * H_ + j] = cc;
            }
        }

        // ---- grid barrier (monotonic counter, one signal per WG per step)
        __threadfence();
        __syncthreads();
        if (tid == 0) {
            __hip_atomic_fetch_add(bar, 1u, __ATOMIC_ACQ_REL,
                                   __HIP_MEMORY_SCOPE_AGENT);
            const unsigned target = (unsigned)(t + 1) * NWG;
            while (__hip_atomic_load(bar, __ATOMIC_ACQUIRE,
                                     __HIP_MEMORY_SCOPE_AGENT) < target) {
                __builtin_amdgcn_s_sleep(1);
            }
        }
        __syncthreads();
        __threadfence();
    }
}

// ---------------------------------------------------------------------------
extern "C" void kernel_launch(void* const* d_in, const int* in_sizes, int n_in,
                              void* d_out, int out_size, void* d_ws, size_t ws_size,
                              hipStream_t stream) {
    const int*   inputs  = (const int*)d_in[0];
    const int*   lengths = (const int*)d_in[1];
    const int*   resets  = (const int*)d_in[2];
    const float* emb     = (const float*)d_in[3];
    const float* W_ih    = (const float*)d_in[4];
    const float* W_hh    = (const float*)d_in[5];
    const float* b_ih    = (const float*)d_in[6];
    const float* b_hh    = (const float*)d_in[7];

    float* out      = (float*)d_out;
    float* out_hT   = out + (size_t)B_ * T_ * H_;
    float* out_cT   = out_hT + B_ * H_;
    float* out_mask = out_cT + B_ * H_;

    char*     ws  = (char*)d_ws;
    unsigned* bar = (unsigned*)ws;               // 256-byte slot
    float*    h0  = (float*)(ws + 256);
    float*    h1  = h0 + B_ * H_;
    float*    c   = h1 + B_ * H_;

    lstm_init<<<(B_ * T_ + 255) / 256, 256, 0, stream>>>(
        lengths, out_mask, h0, h1, c, bar);
    lstm_scan<<<NWG, NTHREADS, 0, stream>>>(
        inputs, lengths, resets, emb, W_ih, W_hh, b_ih, b_hh,
        out, out_hT, out_cT, h0, h1, c, bar);
}